// LenseA_85005992722727
// MI455X (gfx1250) — compile-verified
//
#include <hip/hip_runtime.h>

// ---------------------------------------------------------------------------
// out[b,p,v] = sum_h ( x[b,p,h,:] . W[h,v,:] ) + sum_h bias[h,v]
// == GEMM: C[4096 x 32000] = A[4096 x 4096] * B[4096 x 32000] + biasSum
// Memory-bound: 1.09 GB minimum HBM traffic @ 23.3 TB/s ~= 47 us, far above
// f32-WMMA compute time -> keep f32 precision, optimize data movement:
//   - f32 WMMA (v_wmma_f32_16x16x4_f32), 4x2 16x16 accumulators per wave
//   - grid x walks M so X (64 MB) stays L2-resident; W (512 MB) streams once
//   - async global->LDS (GLOBAL_LOAD_ASYNC_TO_LDS_B128, ASYNCcnt) when the
//     toolchain exposes the builtin; register-staging fallback otherwise
//   - nontemporal stores for Out (write-once, never re-read)
// ---------------------------------------------------------------------------

typedef float v2f __attribute__((ext_vector_type(2)));
typedef float v8f __attribute__((ext_vector_type(8)));
typedef int   v4i_vs __attribute__((vector_size(16)));   // matches builtin sig

#define NHEAD  8
#define DMODEL 512
#define DVOCAB 32000
#define MTOT   4096   // B*P
#define KTOT   4096   // H*D
#define BM     128
#define BN     128
#define KT     32
#define LDSS   36     // padded LDS row stride (floats): 144B rows, 16B aligned,
                      // 36*r mod 64 distinct for r=0..15 -> conflict-free frags

#if __has_builtin(__builtin_amdgcn_global_load_async_to_lds_b128)
#define HAS_ASYNC_LDS 1
#else
#define HAS_ASYNC_LDS 0
#endif

// builtin expects (v4i addrspace(1)*, v4i addrspace(3)*, imm, imm)
#define ASYNC_G(p) ((__attribute__((address_space(1))) v4i_vs*)(p))
#define ASYNC_L(p) ((__attribute__((address_space(3))) v4i_vs*)(p))

__device__ __forceinline__ void wait_async0() {
#if __has_builtin(__builtin_amdgcn_s_wait_asynccnt)
    __builtin_amdgcn_s_wait_asynccnt(0);
#else
    asm volatile("s_wait_asynccnt 0x0" ::: "memory");
#endif
}

__global__ __launch_bounds__(256) void lense_gemm_f32_wmma(
    const float* __restrict__ X,     // [4096][4096]
    const float* __restrict__ W,     // [8][32000][512]
    const float* __restrict__ Bias,  // [8][32000]
    float* __restrict__ Out)         // [4096][32000]
{
    __shared__ float As[2][BM][LDSS];
    __shared__ float Bs[2][BN][LDSS];
    __shared__ float bsum[BN];

    const int tid  = threadIdx.x;
    const int lane = tid & 31;
    const int wave = tid >> 5;
    const int wm   = wave >> 2;      // 0..1  (wave M position, 64 rows each)
    const int wn   = wave & 3;       // 0..3  (wave N position, 32 cols each)
    const int half = lane >> 4;      // lane[4]: selects K pair (A) / M+8 (C)
    const int lrow = lane & 15;      // lane[3:0]

    // blockIdx.x walks M (fast): all 32 M-tiles of one N-tile share the same
    // 2MB W tile through L2 while X stays L2-resident for the whole kernel.
    const int mBase = blockIdx.x * BM;
    const int nBase = blockIdx.y * BN;

    // --- per-block bias column sums: sum_h Bias[h][v] ---
    if (tid < BN) {
        float s = 0.f;
        for (int h = 0; h < NHEAD; ++h)
            s += Bias[(size_t)h * DVOCAB + nBase + tid];
        bsum[tid] = s;
    }

    v8f acc[4][2];
    for (int mi = 0; mi < 4; ++mi)
        for (int ni = 0; ni < 2; ++ni)
            acc[mi][ni] = (v8f)0.f;

    const int aRow = wm * 64;        // + mi*16 + lrow
    const int bRow = wn * 32;        // + ni*16 + lrow

    // tile t: h = t>>4 (8 heads), kt = t&15 (16 K-tiles of 32 per head)
#if HAS_ASYNC_LDS
    // --- async global->LDS fill: 8x b128 per thread, no staging VGPRs ---
    auto issueAsync = [&](int t, int buf) {
        const int h  = t >> 4;
        const int kt = t & 15;
        const float* Ap = X + (size_t)mBase * KTOT + h * DMODEL + kt * KT;
        const float* Bp = W + (size_t)h * DVOCAB * DMODEL
                            + (size_t)nBase * DMODEL + kt * KT;
        #pragma unroll
        for (int i = 0; i < 4; ++i) {
            const int idx = tid + i * 256;
            const int row = idx >> 3;
            const int c4  = (idx & 7) << 2;
            __builtin_amdgcn_global_load_async_to_lds_b128(
                ASYNC_G(Ap + (size_t)row * KTOT + c4),
                ASYNC_L(&As[buf][row][c4]), 0, 0);
            __builtin_amdgcn_global_load_async_to_lds_b128(
                ASYNC_G(Bp + (size_t)row * DMODEL + c4),
                ASYNC_L(&Bs[buf][row][c4]), 0, 0);
        }
    };
#else
    auto ldGlobal = [&](int t, float4 a[4], float4 b[4]) {
        const int h  = t >> 4;
        const int kt = t & 15;
        const float* Ap = X + (size_t)mBase * KTOT + h * DMODEL + kt * KT;
        const float* Bp = W + (size_t)h * DVOCAB * DMODEL
                            + (size_t)nBase * DMODEL + kt * KT;
        #pragma unroll
        for (int i = 0; i < 4; ++i) {
            const int idx = tid + i * 256;
            const int row = idx >> 3;
            const int c4  = (idx & 7) << 2;
            a[i] = *(const float4*)(Ap + (size_t)row * KTOT   + c4);
            b[i] = *(const float4*)(Bp + (size_t)row * DMODEL + c4);
        }
    };
    auto stLds = [&](int buf, const float4 a[4], const float4 b[4]) {
        #pragma unroll
        for (int i = 0; i < 4; ++i) {
            const int idx = tid + i * 256;
            const int row = idx >> 3;
            const int c4  = (idx & 7) << 2;
            *(float4*)&As[buf][row][c4] = a[i];
            *(float4*)&Bs[buf][row][c4] = b[i];
        }
    };
#endif

    // --- compute one KT-deep LDS tile: 8 k-steps of 4, 8 WMMAs per step ---
    auto computeTile = [&](int buf) {
        #pragma unroll
        for (int ks = 0; ks < KT / 4; ++ks) {
            const int kc = ks * 4 + half * 2; // A 16x4: lane[4] picks K pair
            v2f af[4], bf[2];
            #pragma unroll
            for (int mi = 0; mi < 4; ++mi)
                af[mi] = *(const v2f*)&As[buf][aRow + mi * 16 + lrow][kc];
            #pragma unroll
            for (int ni = 0; ni < 2; ++ni)
                bf[ni] = *(const v2f*)&Bs[buf][bRow + ni * 16 + lrow][kc];
            #pragma unroll
            for (int mi = 0; mi < 4; ++mi)
                #pragma unroll
                for (int ni = 0; ni < 2; ++ni)
                    acc[mi][ni] = __builtin_amdgcn_wmma_f32_16x16x4_f32(
                        false, af[mi], false, bf[ni],
                        (short)0, acc[mi][ni], false, false);
        }
    };

    int buf = 0;
#if HAS_ASYNC_LDS
    issueAsync(0, 0);
    wait_async0();
    __syncthreads();
    for (int t = 0; t < 128; ++t) {
        const bool more = (t + 1) < 128;
        if (more) issueAsync(t + 1, buf ^ 1);  // overlap copy with compute
        computeTile(buf);
        if (more) {
            wait_async0();       // this wave's async writes landed in LDS
            __syncthreads();     // all waves' writes visible; prev reads done
            buf ^= 1;
        }
    }
#else
    float4 a4[4], b4[4];
    ldGlobal(0, a4, b4);
    stLds(0, a4, b4);
    __syncthreads();
    for (int t = 0; t < 128; ++t) {
        const bool more = (t + 1) < 128;
        if (more) ldGlobal(t + 1, a4, b4);
        computeTile(buf);
        if (more) {
            stLds(buf ^ 1, a4, b4);
            __syncthreads();
            buf ^= 1;
        }
    }
#endif

    // --- epilogue: C/D layout VGPR r -> M = r + 8*lane[4], N = lane[3:0] ---
    #pragma unroll
    for (int mi = 0; mi < 4; ++mi) {
        #pragma unroll
        for (int ni = 0; ni < 2; ++ni) {
            const int colL = bRow + ni * 16 + lrow;
            const int col  = nBase + colL;
            const float bs = bsum[colL];
            const int row0 = mBase + aRow + mi * 16 + half * 8;
            #pragma unroll
            for (int r = 0; r < 8; ++r) {
                // write-once output: nontemporal so it doesn't evict X from L2
                __builtin_nontemporal_store(
                    acc[mi][ni][r] + bs,
                    Out + (size_t)(row0 + r) * DVOCAB + col);
            }
        }
    }
}

extern "C" void kernel_launch(void* const* d_in, const int* in_sizes, int n_in,
                              void* d_out, int out_size, void* d_ws, size_t ws_size,
                              hipStream_t stream) {
    const float* X    = (const float*)d_in[0];  // [8,512,8,512]
    const float* W    = (const float*)d_in[1];  // [8,32000,512]
    const float* Bias = (const float*)d_in[2];  // [8,32000]
    float* Out        = (float*)d_out;          // [8,512,32000]

    dim3 grid(MTOT / BM, DVOCAB / BN);          // (32, 250); x = M fast
    lense_gemm_f32_wmma<<<grid, 256, 0, stream>>>(X, W, Bias, Out);
}